// Decoder_63307817943739
// MI455X (gfx1250) — compile-verified
//
#include <hip/hip_runtime.h>

#define Bsz 32768
#define Ssz 128
#define Esz 64
#define Hsz 64
#define Tsz 12
#define Fsz 11

#define WAVES_PER_BLOCK 6
#define TILES_TOTAL (Bsz / 16)  // 2048

typedef __attribute__((ext_vector_type(2))) float v2f;
typedef __attribute__((ext_vector_type(8))) float v8f;

#define WMMA4(Af, Bf, Cf) \
  __builtin_amdgcn_wmma_f32_16x16x4_f32(false, (Af), false, (Bf), (short)0, (Cf), false, false)

__device__ __forceinline__ float sigmoidf_(float x) { return 1.0f / (1.0f + expf(-x)); }

// ---------------- Kernel 1: context = mean over S (memory bound, 1.07 GB) ----
__global__ void ctx_mean_k(const float* __restrict__ inp, float* __restrict__ ctx) {
  int gtid = blockIdx.x * blockDim.x + threadIdx.x;
  int b = gtid >> 5;
  int lane = threadIdx.x & 31;
  if (b >= Bsz) return;
  int q = lane & 15;        // which float4 of the 64-float row
  int g = lane >> 4;        // s-phase 0/1
  const float4* p = (const float4*)(inp + (size_t)b * (Ssz * Esz));
  float sx = 0.f, sy = 0.f, sz = 0.f, sw = 0.f;
  for (int s = g; s < Ssz; s += 2) {
    float4 v = p[s * 16 + q];
    sx += v.x; sy += v.y; sz += v.z; sw += v.w;
  }
  sx += __shfl_xor(sx, 16, 32);
  sy += __shfl_xor(sy, 16, 32);
  sz += __shfl_xor(sz, 16, 32);
  sw += __shfl_xor(sw, 16, 32);
  if (g == 0) {
    const float inv = 1.0f / (float)Ssz;
    float4 r; r.x = sx * inv; r.y = sy * inv; r.z = sz * inv; r.w = sw * inv;
    ((float4*)ctx)[b * 16 + q] = r;
  }
}

// ---------------- Kernel 2: LSTM recurrence + predictions via WMMA ----------
__global__ __launch_bounds__(WAVES_PER_BLOCK * 32)
void decoder_k(const float* __restrict__ ctx,     // [B,64]
               const float* __restrict__ y_hists, // [B,12,11]
               const float* __restrict__ y_targs, // [B,5,11]
               const float* __restrict__ fc_w,    // [11,75]
               const float* __restrict__ fc_b,    // [11]
               const float* __restrict__ ffin_w,  // [11,128]
               const float* __restrict__ ffin_b,  // [11]
               const float* __restrict__ w_ih,    // [256,11]
               const float* __restrict__ w_hh,    // [256,64]
               const float* __restrict__ b_ih,    // [256]
               const float* __restrict__ b_hh,    // [256]
               float* __restrict__ out)           // [B,6,11]
{
  __shared__ float WyL[256 * 12];   // w_ih @ fc_w[:,64:75], k=11 padded 0
  __shared__ float WihL[256 * 12];  // w_ih padded, k=11 -> 0
  __shared__ float FcwL[16 * 64];   // fc_w[:, :64], rows >=11 -> 0
  __shared__ float FfinL[16 * 128]; // ffin_w, rows >=11 -> 0
  __shared__ float biasL[256];      // b_ih + b_hh
  __shared__ float fcbL[16];
  __shared__ float ffbL[16];
  __shared__ float hL[WAVES_PER_BLOCK][16 * 64];  // per-wave h transpose buffer

  const int tid = threadIdx.x;
  const int nthreads = blockDim.x;

  // ---- cooperative weight staging (zero-padded, guards OOB fragment reads) ----
  for (int i = tid; i < 256 * 12; i += nthreads) {
    int n = i / 12, k = i % 12;
    WihL[i] = (k < Fsz) ? w_ih[n * Fsz + k] : 0.f;
    float acc = 0.f;
    if (k < Fsz) {
      for (int j = 0; j < Fsz; ++j)
        acc += w_ih[n * Fsz + j] * fc_w[j * (Esz + Fsz) + Esz + k];
    }
    WyL[i] = acc;
  }
  for (int i = tid; i < 16 * 64; i += nthreads) {
    int n = i >> 6, k = i & 63;
    FcwL[i] = (n < Fsz) ? fc_w[n * (Esz + Fsz) + k] : 0.f;
  }
  for (int i = tid; i < 16 * 128; i += nthreads) {
    int n = i >> 7, k = i & 127;
    FfinL[i] = (n < Fsz) ? ffin_w[n * (Hsz + Esz) + k] : 0.f;
  }
  for (int i = tid; i < 256; i += nthreads) biasL[i] = b_ih[i] + b_hh[i];
  if (tid < 16) {
    fcbL[tid] = (tid < Fsz) ? fc_b[tid] : 0.f;
    ffbL[tid] = (tid < Fsz) ? ffin_b[tid] : 0.f;
  }
  __syncthreads();

  const int wid = tid >> 5;
  const int lane = tid & 31;
  const int tile = blockIdx.x * WAVES_PER_BLOCK + wid;
  if (tile >= TILES_TOTAL) return;   // wave-uniform guard

  float* myH = hL[wid];
  const int b0 = tile * 16;
  const int nl = lane & 15;          // M (A) or N (B/C/D) within 16
  const int kh = (lane >> 4) << 1;   // K contribution of lane-half (0 or 2)
  const int mh = (lane >> 4) << 3;   // M offset for C/D rows (0 or 8)

  v2f cpA[3];       // ctx_proj A-fragments (persist across all steps)
  v8f predBase;     // ffin_w[:,64:] @ ctx + ffin_b (D layout)
  {
    // ---- per-tile constants: ctx A-fragments (setup only, dies after) ----
    v2f ctxA[16];
    {
      const float* cp = ctx + (size_t)(b0 + nl) * Esz + kh;
      #pragma unroll
      for (int q = 0; q < 16; ++q) ctxA[q] = *(const v2f*)(cp + q * 4);
    }

    // ctx_proj = fc_w[:, :64] @ ctx + fc_b   (D layout: M=batch, N=f)
    v8f cp8;
    { float bv = fcbL[nl]; cp8 = (v8f){bv, bv, bv, bv, bv, bv, bv, bv}; }
    #pragma unroll
    for (int q = 0; q < 16; ++q) {
      v2f bb = *(const v2f*)(&FcwL[nl * 64 + q * 4 + kh]);
      cp8 = WMMA4(ctxA[q], bb, cp8);
    }
    // transpose ctx_proj (D layout) -> A fragments via LDS [m][12]
    #pragma unroll
    for (int r = 0; r < 8; ++r)
      if (nl < 12) myH[(r + mh) * 12 + nl] = cp8[r];
    #pragma unroll
    for (int q = 0; q < 3; ++q)
      cpA[q] = *(const v2f*)(&myH[nl * 12 + q * 4 + kh]);

    // pred_base
    { float bv = ffbL[nl]; predBase = (v8f){bv, bv, bv, bv, bv, bv, bv, bv}; }
    #pragma unroll
    for (int q = 0; q < 16; ++q) {
      v2f bb = *(const v2f*)(&FfinL[nl * 128 + 64 + q * 4 + kh]);
      predBase = WMMA4(ctxA[q], bb, predBase);
    }
  }

  // ---- init state: h = 0 (in LDS), c = 0 (D-layout regs) ----
  const v8f zero8 = {0.f, 0.f, 0.f, 0.f, 0.f, 0.f, 0.f, 0.f};
  v8f cT[4];
  #pragma unroll
  for (int j = 0; j < 4; ++j) cT[j] = zero8;
  for (int i = lane; i < 16 * 64; i += 32) myH[i] = 0.f;

  // ---- 12 priming + 5 generation steps ----
  #pragma unroll 1
  for (int t = 0; t < 17; ++t) {
    // y A-fragments (rows are 11 floats -> scalar loads; k=11 padded 0)
    const float* yp = (t < Tsz)
        ? (y_hists + ((size_t)(b0 + nl) * Tsz + t) * Fsz)
        : (y_targs + ((size_t)(b0 + nl) * 5 + (t - Tsz)) * Fsz);
    v2f yA[3];
    yA[0].x = yp[kh];     yA[0].y = yp[kh + 1];
    yA[1].x = yp[4 + kh]; yA[1].y = yp[5 + kh];
    if (lane < 16) { yA[2].x = yp[8];  yA[2].y = yp[9]; }
    else           { yA[2].x = yp[10]; yA[2].y = 0.f;   }

    // h A-fragments from LDS
    v2f hA[16];
    #pragma unroll
    for (int q = 0; q < 16; ++q)
      hA[q] = *(const v2f*)(&myH[nl * 64 + q * 4 + kh]);

    // Rolled over j: keeps only acc[4] (32 VGPRs) of accumulators live,
    // holding total VGPR usage under 256 (no s_set_vgpr_msb churn).
    #pragma unroll 1
    for (int j = 0; j < 4; ++j) {       // hidden column group j*16..j*16+15
      v8f acc[4];
      #pragma unroll
      for (int gg = 0; gg < 4; ++gg) {  // i, f, g, o
        const int n = (gg * 4 + j) * 16 + nl;  // gate row 0..255
        float bv = biasL[n];
        v8f a = (v8f){bv, bv, bv, bv, bv, bv, bv, bv};
        const float* whr = w_hh + (size_t)n * Hsz + kh;  // streamed from L2
        #pragma unroll
        for (int q = 0; q < 16; ++q) {
          v2f bb = *(const v2f*)(whr + q * 4);
          a = WMMA4(hA[q], bb, a);
        }
        const float* wir = &WihL[n * 12 + kh];
        const float* wyr = &WyL[n * 12 + kh];
        #pragma unroll
        for (int q = 0; q < 3; ++q) {
          v2f b1 = *(const v2f*)(wir + q * 4);
          a = WMMA4(cpA[q], b1, a);
          v2f b2 = *(const v2f*)(wyr + q * 4);
          a = WMMA4(yA[q], b2, a);
        }
        acc[gg] = a;
      }
      // LSTM cell update (elementwise in D layout), write new h to LDS
      #pragma unroll
      for (int r = 0; r < 8; ++r) {
        float ig = sigmoidf_(acc[0][r]);
        float fg = sigmoidf_(acc[1][r]);
        float gv = tanhf(acc[2][r]);
        float og = sigmoidf_(acc[3][r]);
        float cc = fg * cT[j][r] + ig * gv;
        cT[j][r] = cc;
        myH[(r + mh) * 64 + j * 16 + nl] = og * tanhf(cc);
      }
    }

    // prediction: after last priming step (t==11) and each generation step
    if (t >= 11) {
      v8f p = predBase;
      #pragma unroll
      for (int q = 0; q < 16; ++q) {
        v2f hq = *(const v2f*)(&myH[nl * 64 + q * 4 + kh]);
        v2f bb = *(const v2f*)(&FfinL[nl * 128 + q * 4 + kh]);
        p = WMMA4(hq, bb, p);
      }
      if (nl < Fsz) {
        const int outStep = t - 11;
        #pragma unroll
        for (int r = 0; r < 8; ++r) {
          int bb_ = b0 + r + mh;
          out[(size_t)bb_ * (6 * Fsz) + outStep * Fsz + nl] = p[r];
        }
      }
    }
  }
}

extern "C" void kernel_launch(void* const* d_in, const int* in_sizes, int n_in,
                              void* d_out, int out_size, void* d_ws, size_t ws_size,
                              hipStream_t stream) {
  const float* input_encoded = (const float*)d_in[0];
  const float* y_hists       = (const float*)d_in[1];
  const float* y_targs       = (const float*)d_in[2];
  const float* fc_w          = (const float*)d_in[3];
  const float* fc_b          = (const float*)d_in[4];
  const float* ffin_w        = (const float*)d_in[5];
  const float* ffin_b        = (const float*)d_in[6];
  const float* w_ih          = (const float*)d_in[7];
  const float* w_hh          = (const float*)d_in[8];
  const float* b_ih          = (const float*)d_in[9];
  const float* b_hh          = (const float*)d_in[10];
  (void)in_sizes; (void)n_in; (void)out_size; (void)ws_size;

  float* ctx = (float*)d_ws;         // [B,64] fp32 = 8 MB scratch
  float* out = (float*)d_out;        // [B,6,11] fp32

  // one wave per batch row
  int blocks1 = (Bsz * 32 + 255) / 256;
  ctx_mean_k<<<blocks1, 256, 0, stream>>>(input_encoded, ctx);

  int blocks2 = (TILES_TOTAL + WAVES_PER_BLOCK - 1) / WAVES_PER_BLOCK;
  decoder_k<<<blocks2, WAVES_PER_BLOCK * 32, 0, stream>>>(
      ctx, y_hists, y_targs, fc_w, fc_b, ffin_w, ffin_b,
      w_ih, w_hh, b_ih, b_hh, out);
}